// DenseMRConv_79199196938679
// MI455X (gfx1250) — compile-verified
//
#include <hip/hip_runtime.h>

typedef __attribute__((ext_vector_type(2))) float v2f;
typedef __attribute__((ext_vector_type(4))) float v4f;
typedef __attribute__((ext_vector_type(8))) float v8f;

#define NODES   100000
#define D       64
#define KNB     32
#define TILE    16      // nodes per block
#define HDIM    128     // concat feature dim
#define HSTRIDE 132     // LDS row stride in floats (528B = 33*16: aligned + bank-conflict pad)

__global__ __launch_bounds__(128) void mrconv_fused_wmma(
    const float* __restrict__ x,        // [N, 64]
    const long long* __restrict__ ei,   // [N, 32] int64
    const float* __restrict__ W,        // [128, 64] row-major
    const float* __restrict__ bias,     // [64]
    float* __restrict__ out)            // [N, 64]
{
    __shared__ __align__(16) float     hs[TILE][HSTRIDE];
    __shared__ __align__(16) long long sidx64[TILE * KNB];   // raw int64 indices (4KB)

    const int tid  = threadIdx.x;
    const int base = blockIdx.x * TILE; // first node of this tile

    // ---- stage the 16x32 int64 neighbor indices into LDS via async copy ----
    // 4KB total: 128 threads x 2 x 16B, tracked by ASYNCcnt.
    {
        const unsigned long long src =
            (unsigned long long)(uintptr_t)(ei + (size_t)base * KNB);
        const unsigned lds0 = (unsigned)(uintptr_t)(&sidx64[0]);  // low 32 bits = LDS offset
        #pragma unroll
        for (int i = 0; i < 2; ++i) {
            const unsigned off = (unsigned)(tid * 16 + i * 2048);
            asm volatile(
                "global_load_async_to_lds_b128 %0, %1, %2"
                :
                : "v"(lds0 + off), "v"(off), "s"(src)
                : "memory");
        }
        asm volatile("s_wait_asynccnt 0x0" ::: "memory");
    }
    __syncthreads();
    const int* sidx = (const int*)sidx64;   // low dword of each little-endian int64

    // ---- phase 1: build h = [x_i | max_k(x_j) - x_i] tile in LDS ----
    // Half-wave covers one node row: lane -> float4 of features. b128 gathers (L2 hits).
    const int lane = tid & 31;
    const int wave = tid >> 5;
    const int sub  = lane >> 4;          // which half-wave
    const int fq   = (lane & 15) * 4;    // feature quad [fq..fq+3]

    #pragma unroll
    for (int p = 0; p < 2; ++p) {
        const int n = wave * 4 + p * 2 + sub;   // node within tile (each wave: 4 nodes)
        const int g = base + n;
        const v4f xi = *(const v4f*)(x + g * D + fq);
        v4f m = { -3.402823466e+38f, -3.402823466e+38f,
                  -3.402823466e+38f, -3.402823466e+38f };
        #pragma unroll 8
        for (int k = 0; k < KNB; ++k) {
            const int idx = sidx[(n * KNB + k) * 2];      // broadcast read per half-wave
            const v4f xj = *(const v4f*)(x + idx * D + fq); // global_load_b128, L2 hit
            m.x = fmaxf(m.x, xj.x);
            m.y = fmaxf(m.y, xj.y);
            m.z = fmaxf(m.z, xj.z);
            m.w = fmaxf(m.w, xj.w);
        }
        // max_k(x_j - x_i) == max_k(x_j) - x_i : exact (fp rounding is monotonic)
        *(v4f*)(&hs[n][fq]) = xi;
        v4f dm = { m.x - xi.x, m.y - xi.y, m.z - xi.z, m.w - xi.w };
        *(v4f*)(&hs[n][D + fq]) = dm;
    }
    __syncthreads();

    // ---- phase 2: 16x16 output tile per wave via V_WMMA_F32_16X16X4_F32 ----
    const int row     = lane & 15;        // M (A/C) or N (B/C)
    const int ksel    = (lane >> 4) << 1; // lanes 16-31 carry K+2,K+3
    const int colbase = wave * 16;

    v8f c = {};                           // f32 accumulator, starts at 0
    #pragma unroll
    for (int k0 = 0; k0 < HDIM; k0 += 4) {
        v2f a;                            // A 16x4 fragment from LDS (row = M)
        a.x = hs[row][k0 + ksel];
        a.y = hs[row][k0 + ksel + 1];
        v2f bfrag;                        // B 4x16 fragment from W (row-major [128,64])
        bfrag.x = W[(k0 + ksel)     * D + colbase + row];
        bfrag.y = W[(k0 + ksel + 1) * D + colbase + row];
        c = __builtin_amdgcn_wmma_f32_16x16x4_f32(
                /*neg_a=*/false, a, /*neg_b=*/false, bfrag,
                /*c_mod=*/(short)0, c, /*reuse_a=*/false, /*reuse_b=*/false);
    }

    // ---- epilogue: bias + store (C layout: VGPR r -> M=r (lanes 0-15), M=r+8) ----
    const float bv   = bias[colbase + row];
    const int   rofs = (lane >> 4) << 3;  // 0 or 8
    #pragma unroll
    for (int r = 0; r < 8; ++r) {
        const int m = r + rofs;
        out[(base + m) * D + colbase + row] = c[r] + bv;
    }
}

extern "C" void kernel_launch(void* const* d_in, const int* in_sizes, int n_in,
                              void* d_out, int out_size, void* d_ws, size_t ws_size,
                              hipStream_t stream) {
    const float*     x  = (const float*)d_in[0];
    const long long* ei = (const long long*)d_in[1];   // int64 per reference
    const float*     W  = (const float*)d_in[2];
    const float*     b  = (const float*)d_in[3];
    float*           o  = (float*)d_out;

    dim3 grid(NODES / TILE);   // 6250 tiles, exact
    dim3 block(128);           // 4 wave32s
    mrconv_fused_wmma<<<grid, block, 0, stream>>>(x, ei, W, b, o);
}